// PF_28192165331387
// MI455X (gfx1250) — compile-verified
//
#include <hip/hip_runtime.h>
#include <hip/hip_bf16.h>
#include <math.h>

typedef float v2f __attribute__((ext_vector_type(2)));
typedef float v8f __attribute__((ext_vector_type(8)));

#define NPART  1000000
#define NTILES (NPART / 16)

// ---- workspace layout (float-element offsets into d_ws) ----
#define OFF_TT    0      // Tt[k][j] = T[j][k]      (256)
#define OFF_VT    256    // Vt[k][j] = V[j][k] pad0 (256)
#define OFF_W     512    // w[16]
#define OFF_Z0    528    // z0 padded to 16
#define OFF_CONST 544    // [0]=log_norm, [1]=qtotal
#define OFF_G     560    // G = I-KC (256)
#define OFF_MEAN  816    // sum of c_p * x_p (16)
#define OFF_COV   832    // sum of c_p * x_p x_p^T (256)
#define OFF_BSUM  1088   // scan block sums (<=4096)
#define OFF_XPOST 8192L
#define OFF_Q     (OFF_XPOST + (long)NPART * 16)
#define OFF_CUM   (OFF_Q + (long)NPART)
#define OFF_CNT   (OFF_CUM + (long)NPART)

// ---- counter-based RNG (deterministic across calls) ----
__device__ __forceinline__ unsigned pcg(unsigned v) {
  v = v * 747796405u + 2891336453u;
  unsigned w = ((v >> ((v >> 28) + 4u)) ^ v) * 277803737u;
  return (w >> 22) ^ w;
}
__device__ __forceinline__ float u01(unsigned h) {
  return ((float)(h >> 8) + 0.5f) * (1.0f / 16777216.0f);   // in (0,1)
}
__device__ __forceinline__ float gaussn(int p, int k) {
  unsigned s = (unsigned)(p * 16 + k);
  float u1 = u01(pcg(s ^ 0x68bc21ebu));
  float u2 = u01(pcg(s + 0x02e5be93u));
  return __fsqrt_rn(-2.0f * __logf(u1)) * __cosf(6.28318530718f * u2);
}

// =====================================================================
// Kernel 1: fold all small linear algebra into T, V, w, z0, G, log_norm
// =====================================================================
__global__ void pf_setup(const float* __restrict__ x, const float* __restrict__ y,
                         const float* __restrict__ u, const float* __restrict__ P,
                         const float* __restrict__ Q, const float* __restrict__ R,
                         const float* __restrict__ A, const float* __restrict__ B,
                         const float* __restrict__ C, const float* __restrict__ D,
                         float* __restrict__ ws) {
  if (threadIdx.x != 0 || blockIdx.x != 0) return;

  float Lq[256], Lr[64];
  for (int i = 0; i < 256; ++i) Lq[i] = 0.f;
  for (int j = 0; j < 16; ++j) {
    float s = Q[j * 16 + j];
    for (int k = 0; k < j; ++k) s -= Lq[j * 16 + k] * Lq[j * 16 + k];
    float d = __fsqrt_rn(fmaxf(s, 1e-20f));
    Lq[j * 16 + j] = d;
    for (int i = j + 1; i < 16; ++i) {
      float t = Q[i * 16 + j];
      for (int k = 0; k < j; ++k) t -= Lq[i * 16 + k] * Lq[j * 16 + k];
      Lq[i * 16 + j] = t / d;
    }
  }
  for (int i = 0; i < 64; ++i) Lr[i] = 0.f;
  for (int j = 0; j < 8; ++j) {
    float s = R[j * 8 + j];
    for (int k = 0; k < j; ++k) s -= Lr[j * 8 + k] * Lr[j * 8 + k];
    float d = __fsqrt_rn(fmaxf(s, 1e-20f));
    Lr[j * 8 + j] = d;
    for (int i = j + 1; i < 8; ++i) {
      float t = R[i * 8 + j];
      for (int k = 0; k < j; ++k) t -= Lr[i * 8 + k] * Lr[j * 8 + k];
      Lr[i * 8 + j] = t / d;
    }
  }
  float logdet = 0.f;
  for (int j = 0; j < 8; ++j) logdet += __logf(Lr[j * 8 + j]);
  logdet *= 2.f;
  const float log_norm = -0.5f * (8.f * 1.8378770664093f + logdet);

  // P_prior = A P A^T + Q
  float AP[256], Pp[256];
  for (int i = 0; i < 16; ++i)
    for (int k = 0; k < 16; ++k) {
      float s = 0.f;
      for (int j = 0; j < 16; ++j) s += A[i * 16 + j] * P[j * 16 + k];
      AP[i * 16 + k] = s;
    }
  for (int i = 0; i < 16; ++i)
    for (int k = 0; k < 16; ++k) {
      float s = Q[i * 16 + k];
      for (int j = 0; j < 16; ++j) s += AP[i * 16 + j] * A[k * 16 + j];
      Pp[i * 16 + k] = s;
    }
  // S = C P_prior C^T + R
  float CP[128], S[64];
  for (int j = 0; j < 8; ++j)
    for (int k = 0; k < 16; ++k) {
      float s = 0.f;
      for (int i = 0; i < 16; ++i) s += C[j * 16 + i] * Pp[i * 16 + k];
      CP[j * 16 + k] = s;
    }
  for (int j = 0; j < 8; ++j)
    for (int l = 0; l < 8; ++l) {
      float s = R[j * 8 + l];
      for (int k = 0; k < 16; ++k) s += CP[j * 16 + k] * C[l * 16 + k];
      S[j * 8 + l] = s;
    }
  // Sinv via Cholesky (S SPD => pinv == inv)
  float Ls[64], Li[64], Sinv[64];
  for (int i = 0; i < 64; ++i) { Ls[i] = 0.f; Li[i] = 0.f; }
  for (int j = 0; j < 8; ++j) {
    float s = S[j * 8 + j];
    for (int k = 0; k < j; ++k) s -= Ls[j * 8 + k] * Ls[j * 8 + k];
    float d = __fsqrt_rn(fmaxf(s, 1e-20f));
    Ls[j * 8 + j] = d;
    for (int i = j + 1; i < 8; ++i) {
      float t = S[i * 8 + j];
      for (int k = 0; k < j; ++k) t -= Ls[i * 8 + k] * Ls[j * 8 + k];
      Ls[i * 8 + j] = t / d;
    }
  }
  for (int c = 0; c < 8; ++c) {
    float z[8];
    for (int i = 0; i < 8; ++i) z[i] = 0.f;
    for (int i = c; i < 8; ++i) {
      float s = (i == c) ? 1.f : 0.f;
      for (int k = c; k < i; ++k) s -= Ls[i * 8 + k] * z[k];
      z[i] = s / Ls[i * 8 + i];
      Li[i * 8 + c] = z[i];
    }
  }
  for (int a = 0; a < 8; ++a)
    for (int b = 0; b < 8; ++b) {
      float s = 0.f;
      for (int i = 0; i < 8; ++i) s += Li[i * 8 + a] * Li[i * 8 + b];
      Sinv[a * 8 + b] = s;
    }
  // K = P_prior C^T Sinv
  float PCt[128], K[128];
  for (int i = 0; i < 16; ++i)
    for (int j = 0; j < 8; ++j) {
      float s = 0.f;
      for (int k = 0; k < 16; ++k) s += Pp[i * 16 + k] * C[j * 16 + k];
      PCt[i * 8 + j] = s;
    }
  for (int i = 0; i < 16; ++i)
    for (int j = 0; j < 8; ++j) {
      float s = 0.f;
      for (int l = 0; l < 8; ++l) s += PCt[i * 8 + l] * Sinv[l * 8 + j];
      K[i * 8 + j] = s;
    }
  // G = I - K C ; ALq = A Lq ; T = G ALq
  float G[256], ALq[256], T[256];
  for (int i = 0; i < 16; ++i)
    for (int k = 0; k < 16; ++k) {
      float s = (i == k) ? 1.f : 0.f;
      for (int j = 0; j < 8; ++j) s -= K[i * 8 + j] * C[j * 16 + k];
      G[i * 16 + k] = s;
    }
  for (int i = 0; i < 16; ++i)
    for (int j = 0; j < 16; ++j) {
      float s = 0.f;
      for (int k = 0; k < 16; ++k) s += A[i * 16 + k] * Lq[k * 16 + j];
      ALq[i * 16 + j] = s;
    }
  for (int i = 0; i < 16; ++i)
    for (int j = 0; j < 16; ++j) {
      float s = 0.f;
      for (int k = 0; k < 16; ++k) s += G[i * 16 + k] * ALq[k * 16 + j];
      T[i * 16 + j] = s;
    }
  // w = G(Ax+Bu) + K(y-Du) ; d0 = C(Ax+Bu)+Du-y
  float xu[16], ymDu[8], w[16], d0[8];
  for (int i = 0; i < 16; ++i) {
    float s = 0.f;
    for (int k = 0; k < 16; ++k) s += A[i * 16 + k] * x[k];
    for (int l = 0; l < 8; ++l) s += B[i * 8 + l] * u[l];
    xu[i] = s;
  }
  for (int j = 0; j < 8; ++j) {
    float s = y[j];
    for (int l = 0; l < 8; ++l) s -= D[j * 8 + l] * u[l];
    ymDu[j] = s;
  }
  for (int i = 0; i < 16; ++i) {
    float s = 0.f;
    for (int k = 0; k < 16; ++k) s += G[i * 16 + k] * xu[k];
    for (int j = 0; j < 8; ++j) s += K[i * 8 + j] * ymDu[j];
    w[i] = s;
  }
  for (int j = 0; j < 8; ++j) {
    float s = -y[j];
    for (int k = 0; k < 16; ++k) s += C[j * 16 + k] * xu[k];
    for (int l = 0; l < 8; ++l) s += D[j * 8 + l] * u[l];
    d0[j] = s;
  }
  // U = C ALq ; V = Lr^{-1} U ; z0 = Lr^{-1} d0
  float U8[128], V[128], z0[8];
  for (int j = 0; j < 8; ++j)
    for (int k = 0; k < 16; ++k) {
      float s = 0.f;
      for (int i = 0; i < 16; ++i) s += C[j * 16 + i] * ALq[i * 16 + k];
      U8[j * 16 + k] = s;
    }
  for (int i = 0; i < 8; ++i) {
    float s = d0[i];
    for (int k = 0; k < i; ++k) s -= Lr[i * 8 + k] * z0[k];
    z0[i] = s / Lr[i * 8 + i];
  }
  for (int k = 0; k < 16; ++k) {
    float z[8];
    for (int i = 0; i < 8; ++i) {
      float s = U8[i * 16 + k];
      for (int t = 0; t < i; ++t) s -= Lr[i * 8 + t] * z[t];
      z[i] = s / Lr[i * 8 + i];
      V[i * 16 + k] = z[i];
    }
  }
  // publish
  for (int k = 0; k < 16; ++k)
    for (int j = 0; j < 16; ++j) {
      ws[OFF_TT + k * 16 + j] = T[j * 16 + k];
      ws[OFF_VT + k * 16 + j] = (j < 8) ? V[j * 16 + k] : 0.f;
    }
  for (int i = 0; i < 16; ++i) {
    ws[OFF_W + i] = w[i];
    ws[OFF_Z0 + i] = (i < 8) ? z0[i] : 0.f;
  }
  for (int i = 0; i < 256; ++i) ws[OFF_G + i] = G[i];
  ws[OFF_CONST + 0] = log_norm;
}

// =====================================================================
// Kernel 2: per-particle WMMA transform + likelihood
//   x_post = w + T n ;  q = exp(log_norm - 0.5*||V n + z0||^2) + 0.01
// =====================================================================
__global__ void __launch_bounds__(256) pf_particles(const float* __restrict__ ws,
                                                    float* __restrict__ xpost,
                                                    float* __restrict__ qarr) {
  __shared__ float sTt[256], sVt[256], sW[16], sZ0[16], sLn[1];
  for (int i = threadIdx.x; i < 256; i += 256) { sTt[i] = ws[OFF_TT + i]; sVt[i] = ws[OFF_VT + i]; }
  if (threadIdx.x < 16) { sW[threadIdx.x] = ws[OFF_W + threadIdx.x]; sZ0[threadIdx.x] = ws[OFF_Z0 + threadIdx.x]; }
  if (threadIdx.x == 0) sLn[0] = ws[OFF_CONST];
  __syncthreads();

  const int lane = threadIdx.x & 31;
  const int col  = lane & 15;       // D-matrix column (state dim / z dim)
  const int half = lane >> 4;       // 0: rows M0..7 ; 1: rows M8..15
  const int wave = blockIdx.x * (blockDim.x >> 5) + (threadIdx.x >> 5);
  const int nw   = gridDim.x * (blockDim.x >> 5);
  const float ln = sLn[0];

  for (int t = wave; t < NTILES; t += nw) {
    const int pbase = t * 16;
    const int p = pbase + col;      // A operand: lane = particle row M
    v8f accX = {};
    v8f accZ = {};
#pragma unroll
    for (int c = 0; c < 4; ++c) {
      const int kb = 4 * c + 2 * half;              // K chunk split across halves
      v2f a; a.x = gaussn(p, kb); a.y = gaussn(p, kb + 1);
      v2f bx; bx.x = sTt[kb * 16 + col]; bx.y = sTt[(kb + 1) * 16 + col];
      v2f bz; bz.x = sVt[kb * 16 + col]; bz.y = sVt[(kb + 1) * 16 + col];
      accX = __builtin_amdgcn_wmma_f32_16x16x4_f32(false, a, false, bx, (short)0, accX, false, false);
      accZ = __builtin_amdgcn_wmma_f32_16x16x4_f32(false, a, false, bz, (short)0, accZ, false, false);
    }
    const float wadd = sW[col];
    const float z0v  = sZ0[col];
#pragma unroll
    for (int r = 0; r < 8; ++r) {
      const int row = pbase + r + 8 * half;
      __builtin_nontemporal_store(accX[r] + wadd, &xpost[(size_t)row * 16 + col]);
      float zv = accZ[r] + z0v;
      float sq = (col < 8) ? zv * zv : 0.f;
      sq += __shfl_xor(sq, 1, 16);
      sq += __shfl_xor(sq, 2, 16);
      sq += __shfl_xor(sq, 4, 16);
      sq += __shfl_xor(sq, 8, 16);
      if (col == 0) __builtin_nontemporal_store(__expf(ln - 0.5f * sq) + 0.01f, &qarr[row]);
    }
  }
}

// =====================================================================
// Kernels 3-5: inclusive scan of q (cumsum) + total
// =====================================================================
__global__ void __launch_bounds__(256) pf_scan1(const float* __restrict__ q,
                                                float* __restrict__ cum,
                                                float* __restrict__ bsum) {
  __shared__ float sh[256];
  const int t = threadIdx.x;
  const long base = (long)blockIdx.x * 4096 + (long)t * 16;
  float v[16];
  float s = 0.f;
#pragma unroll
  for (int i = 0; i < 16; ++i) {
    long idx = base + i;
    float xv = (idx < NPART) ? q[idx] : 0.f;
    v[i] = xv; s += xv;
  }
  sh[t] = s; __syncthreads();
  for (int off = 1; off < 256; off <<= 1) {
    float add = (t >= off) ? sh[t - off] : 0.f;
    __syncthreads();
    sh[t] += add;
    __syncthreads();
  }
  float run = sh[t] - s;     // exclusive prefix for this thread
#pragma unroll
  for (int i = 0; i < 16; ++i) {
    long idx = base + i;
    if (idx < NPART) { run += v[i]; cum[idx] = run; }
  }
  if (t == 255) bsum[blockIdx.x] = sh[255];
}

__global__ void __launch_bounds__(1024) pf_scan2(float* __restrict__ bsum,
                                                 float* __restrict__ ws, int nb) {
  __shared__ float sh[1024];
  const int t = threadIdx.x;
  float v = (t < nb) ? bsum[t] : 0.f;
  sh[t] = v; __syncthreads();
  for (int off = 1; off < 1024; off <<= 1) {
    float add = (t >= off) ? sh[t - off] : 0.f;
    __syncthreads();
    sh[t] += add;
    __syncthreads();
  }
  if (t < nb) bsum[t] = sh[t] - v;           // exclusive block offsets
  if (t == 1023) ws[OFF_CONST + 1] = sh[1023];  // grand total
}

__global__ void __launch_bounds__(256) pf_scan3(float* __restrict__ cum,
                                                const float* __restrict__ bsum) {
  int i = blockIdx.x * blockDim.x + threadIdx.x;
  if (i < NPART) cum[i] += bsum[i >> 12];
}

// =====================================================================
// Kernel 6: systematic-style resampling via searchsorted(left) + counts
// =====================================================================
__global__ void __launch_bounds__(256) pf_resample(const float* __restrict__ cum,
                                                   const float* __restrict__ ws,
                                                   int* __restrict__ counts) {
  const float tot = ws[OFF_CONST + 1];
  const int stride = gridDim.x * blockDim.x;
  for (int i = blockIdx.x * blockDim.x + threadIdx.x; i < NPART; i += stride) {
    float r = u01(pcg((unsigned)i ^ 0x85ebca6bu)) * tot;
    int lo = 0, hi = NPART;
    while (lo < hi) {                 // first j with cum[j] >= r
      int mid = (lo + hi) >> 1;
      if (cum[mid] < r) lo = mid + 1; else hi = mid;
    }
    int idx = (lo < NPART - 1) ? lo : (NPART - 1);
    atomicAdd(&counts[idx], 1);
  }
}

// =====================================================================
// Kernel 7: count-weighted mean + WMMA outer-product covariance
//   mean_acc  += sum c_p x_p     cov_acc += sum c_p x_p x_p^T
// =====================================================================
__global__ void __launch_bounds__(256) pf_meancov(const float* __restrict__ xpost,
                                                  const int* __restrict__ counts,
                                                  float* __restrict__ ws) {
  __shared__ float shm[16];
  if (threadIdx.x < 16) shm[threadIdx.x] = 0.f;
  __syncthreads();

  // weighted mean
  const int gid = blockIdx.x * blockDim.x + threadIdx.x;
  const int gstride = gridDim.x * blockDim.x;
  float loc[16];
#pragma unroll
  for (int j = 0; j < 16; ++j) loc[j] = 0.f;
  for (int p = gid; p < NPART; p += gstride) {
    int c = counts[p];
    if (c) {
      float fc = (float)c;
#pragma unroll
      for (int j = 0; j < 16; ++j) loc[j] += fc * xpost[(size_t)p * 16 + j];
    }
  }
#pragma unroll
  for (int j = 0; j < 16; ++j) atomicAdd(&shm[j], loc[j]);
  __syncthreads();
  if (threadIdx.x < 16) atomicAdd(&ws[OFF_MEAN + threadIdx.x], shm[threadIdx.x]);
  __syncthreads();

  // weighted second moment via WMMA: D[i][j] += sum_p (c_p x[p,i]) * x[p,j]
  const int lane = threadIdx.x & 31;
  const int col  = lane & 15;
  const int half = lane >> 4;
  const int wave = blockIdx.x * (blockDim.x >> 5) + (threadIdx.x >> 5);
  const int nw   = gridDim.x * (blockDim.x >> 5);
  v8f acc = {};
  for (int t = wave; t < NTILES; t += nw) {
    const int pbase = t * 16;
    if (t + nw < NTILES)
      __builtin_prefetch(&xpost[(size_t)(pbase + 16 * nw) * 16], 0, 1);
#pragma unroll
    for (int c = 0; c < 4; ++c) {
      const int p0 = pbase + 4 * c + 2 * half;
      const float x0 = xpost[(size_t)p0 * 16 + col];
      const float x1 = xpost[(size_t)(p0 + 1) * 16 + col];
      v2f a; a.x = (float)counts[p0] * x0; a.y = (float)counts[p0 + 1] * x1;
      v2f b; b.x = x0; b.y = x1;
      acc = __builtin_amdgcn_wmma_f32_16x16x4_f32(false, a, false, b, (short)0, acc, false, false);
    }
  }
#pragma unroll
  for (int r = 0; r < 8; ++r)
    atomicAdd(&ws[OFF_COV + (size_t)(r + 8 * half) * 16 + col], acc[r]);
}

// =====================================================================
// Kernel 8: finalize — P_post from moments, P_out = G P_post, outputs
// =====================================================================
__global__ void __launch_bounds__(256) pf_finalize(const float* __restrict__ ws,
                                                   float* __restrict__ out) {
  __shared__ float Pp[256], mean[16];
  const int t = threadIdx.x;
  if (t < 16) mean[t] = ws[OFF_MEAN + t] * (1.0f / (float)NPART);
  __syncthreads();
  const int i = t >> 4, j = t & 15;
  float cov = ws[OFF_COV + t];
  Pp[t] = (cov - (float)NPART * mean[i] * mean[j]) * (1.0f / (float)(NPART - 1));
  __syncthreads();
  float s = 0.f;
#pragma unroll
  for (int k = 0; k < 16; ++k) s += ws[OFF_G + i * 16 + k] * Pp[k * 16 + j];
  out[16 + t] = s;
  if (t < 16) out[t] = mean[t];
}

// =====================================================================
extern "C" void kernel_launch(void* const* d_in, const int* in_sizes, int n_in,
                              void* d_out, int out_size, void* d_ws, size_t ws_size,
                              hipStream_t stream) {
  const float* x = (const float*)d_in[0];
  const float* y = (const float*)d_in[1];
  const float* u = (const float*)d_in[2];
  const float* P = (const float*)d_in[3];
  const float* Q = (const float*)d_in[4];
  const float* R = (const float*)d_in[5];
  const float* A = (const float*)d_in[6];
  const float* B = (const float*)d_in[7];
  const float* C = (const float*)d_in[8];
  const float* D = (const float*)d_in[9];

  float* ws    = (float*)d_ws;
  float* xpost = ws + OFF_XPOST;
  float* qarr  = ws + OFF_Q;
  float* cum   = ws + OFF_CUM;
  int*   counts = (int*)(ws + OFF_CNT);

  hipMemsetAsync(ws + OFF_MEAN, 0, (16 + 256) * sizeof(float), stream);
  hipMemsetAsync(counts, 0, (size_t)NPART * sizeof(int), stream);

  pf_setup<<<1, 32, 0, stream>>>(x, y, u, P, Q, R, A, B, C, D, ws);
  pf_particles<<<512, 256, 0, stream>>>(ws, xpost, qarr);

  const int nScanBlocks = (NPART + 4095) / 4096;   // 245
  pf_scan1<<<nScanBlocks, 256, 0, stream>>>(qarr, cum, ws + OFF_BSUM);
  pf_scan2<<<1, 1024, 0, stream>>>(ws + OFF_BSUM, ws, nScanBlocks);
  pf_scan3<<<(NPART + 255) / 256, 256, 0, stream>>>(cum, ws + OFF_BSUM);

  pf_resample<<<2048, 256, 0, stream>>>(cum, ws, counts);
  pf_meancov<<<512, 256, 0, stream>>>(xpost, counts, ws);
  pf_finalize<<<1, 256, 0, stream>>>(ws, (float*)d_out);
}